// TripletLossAttribute_23716809408617
// MI455X (gfx1250) — compile-verified
//
#include <hip/hip_runtime.h>
#include <math.h>

// CDNA5 / gfx1250, wave32. One fused kernel, one workgroup (16 waves = 512 thr).
// Per-group Gram matrices via V_WMMA_F32_16X16X4_F32, fused weighted-distance
// combine, masked row max/min butterfly reduction, scalar hinge-loss output.

typedef __attribute__((ext_vector_type(2))) float v2f;
typedef __attribute__((ext_vector_type(8))) float v8f;

#define NROW 64
#define KDIM 2048
#define NG   6

__device__ const int kGroupStart[NG + 1] = {0, 341, 682, 1023, 1364, 1705, 2048};

__device__ __forceinline__ v2f ld2(const float* __restrict__ p) {
    v2f v;
    v.x = p[0];
    v.y = p[1];
    return v;
}

__global__ __launch_bounds__(512)
void triplet_attr_kernel(const float* __restrict__ X,     // [64][2048]
                         const int*   __restrict__ tgt,   // [64]
                         const float* __restrict__ att,   // [64][64][6]
                         float*       __restrict__ out)   // [1]
{
    __shared__ float n_lds[NROW][NG];   // per-row per-group squared norms
    __shared__ int   tgt_lds[NROW];
    __shared__ float part_ap[NROW][4];
    __shared__ float part_an[NROW][4];
    __shared__ float loss_lds[NROW];

    const int tid = threadIdx.x;

    // ---- Phase 0: per-row, per-group squared norms (384 threads) ----
    if (tid < NROW * NG) {
        const int row = tid / NG;
        const int g   = tid % NG;
        const int ks  = kGroupStart[g];
        const int ke  = kGroupStart[g + 1];
        const float* xr = X + row * KDIM;
        float s = 0.0f;
        for (int k = ks; k < ke; ++k) {
            float v = xr[k];
            s = fmaf(v, v, s);
        }
        n_lds[row][g] = s;
    }
    if (tid < NROW) tgt_lds[tid] = tgt[tid];
    __syncthreads();

    // ---- Phase 1: each wave owns one 16x16 tile of the 64x64 sq matrix ----
    const int wave = tid >> 5;          // 0..15
    const int lane = tid & 31;
    const int half = lane >> 4;         // 0: K+0/1 (A), M=r (C) ; 1: K+2/3, M=r+8
    const int l16  = lane & 15;
    const int ti   = wave >> 2;         // row-tile (i)
    const int tj   = wave & 3;          // col-tile (j)
    const int rowA = ti * 16 + l16;     // A-matrix row (M)
    const int rowB = tj * 16 + l16;     // B-matrix column (N)
    const int j    = tj * 16 + l16;     // this lane's output column

    v8f sq = {};                        // weighted squared-distance accumulator

    for (int g = 0; g < NG; ++g) {
        const int ks    = kGroupStart[g];
        const int ke    = kGroupStart[g + 1];
        const int nfull = (ke - ks) >> 2;            // full 4-wide K steps (85)
        const float* pa = X + rowA * KDIM + ks + half * 2;
        const float* pb = X + rowB * KDIM + ks + half * 2;

        v8f c = {};                     // Gram tile accumulator for group g

        // Branch-free main loop, software-pipelined one stage:
        // loads for step t+1 are in flight while step t's WMMA executes.
        v2f a = ld2(pa);
        v2f b = ld2(pb);
        for (int t = 1; t < nfull; ++t) {
            v2f a2 = ld2(pa + 4 * t);
            v2f b2 = ld2(pb + 4 * t);
            c = __builtin_amdgcn_wmma_f32_16x16x4_f32(
                    false, a, false, b, (short)0, c, false, false);
            a = a2;
            b = b2;
        }
        c = __builtin_amdgcn_wmma_f32_16x16x4_f32(
                false, a, false, b, (short)0, c, false, false);

        // Group remainder (1 or 3 K values): one predicated step, zero-filled.
        {
            const int kk = ks + nfull * 4 + (half ? 2 : 0);
            float ax = 0.0f, ay = 0.0f, bx = 0.0f, by = 0.0f;
            if (kk < ke) {
                ax = X[rowA * KDIM + kk];
                bx = X[rowB * KDIM + kk];
            }
            if (kk + 1 < ke) {
                ay = X[rowA * KDIM + kk + 1];
                by = X[rowB * KDIM + kk + 1];
            }
            v2f at = {ax, ay};
            v2f bt = {bx, by};
            c = __builtin_amdgcn_wmma_f32_16x16x4_f32(
                    false, at, false, bt, (short)0, c, false, false);
        }

        // Fuse: sq[i,j] += att[i,j,g]^2 * (n_i^g + n_j^g - 2*G_g[i,j])
        #pragma unroll
        for (int r = 0; r < 8; ++r) {
            const int i   = ti * 16 + r + half * 8;
            const float aw = att[(i * 64 + j) * NG + g];
            sq[r] += aw * aw * (n_lds[i][g] + n_lds[j][g] - 2.0f * c[r]);
        }
    }

    // ---- Phase 2: dist + masked row max/min within this tile ----
    const float INF = __builtin_inff();
    const int tgt_j = tgt_lds[j];
    #pragma unroll
    for (int r = 0; r < 8; ++r) {
        const int i  = ti * 16 + r + half * 8;
        const float d = sqrtf(fmaxf(sq[r], 1e-12f));
        const bool same = (tgt_lds[i] == tgt_j);
        float vap = same ? d : -INF;    // candidate for dist_ap (max over same)
        float van = same ? INF : d;     // candidate for dist_an (min over diff)
        // Butterfly over the 16 columns of this tile (stays within 16-lane half)
        #pragma unroll
        for (int m = 1; m < 16; m <<= 1) {
            vap = fmaxf(vap, __shfl_xor(vap, m, 32));
            van = fminf(van, __shfl_xor(van, m, 32));
        }
        if (l16 == 0) {
            part_ap[i][tj] = vap;
            part_an[i][tj] = van;
        }
    }
    __syncthreads();

    // ---- Phase 3: finish across the 4 column tiles, per-row hinge ----
    if (tid < NROW) {
        float ap = -INF, an = INF;
        #pragma unroll
        for (int t = 0; t < 4; ++t) {
            ap = fmaxf(ap, part_ap[tid][t]);
            an = fminf(an, part_an[tid][t]);
        }
        loss_lds[tid] = fmaxf(ap - an + 0.3f, 0.0f);
    }
    __syncthreads();

    if (tid == 0) {
        float s = 0.0f;
        for (int r = 0; r < NROW; ++r) s += loss_lds[r];
        out[0] = s * (1.0f / NROW);
    }
}

extern "C" void kernel_launch(void* const* d_in, const int* in_sizes, int n_in,
                              void* d_out, int out_size, void* d_ws, size_t ws_size,
                              hipStream_t stream) {
    const float* X   = (const float*)d_in[0];   // inputs (64,2048) fp32
    const int*   tgt = (const int*)d_in[1];     // targets (64) int
    const float* att = (const float*)d_in[2];   // attention_attribute (64,64,6) fp32
    float* out = (float*)d_out;
    triplet_attr_kernel<<<1, 512, 0, stream>>>(X, tgt, att, out);
}